// TextModel_hidden_67113158967874
// MI455X (gfx1250) — compile-verified
//
#include <hip/hip_runtime.h>
#include <hip/hip_bf16.h>

// ---------------------------------------------------------------------------
// TextModel (embedding -> LSTM(300->50) -> LSTM(50->10) -> FC -> FC) for
// MI455X (gfx1250, wave32).
//   K1: fused embedding-gather + input-projection GEMM  [T*B,300]x[300,200]
//       via V_WMMA_F32_16X16X4_F32; 7 N-tiles per wave so each gathered
//       A float2 feeds 7 wmma (branch-free inner loop).
//   K2: persistent LSTM1 scan; h@W_hh^T via WMMA from LDS, accumulator
//       seeded from precomputed input gates; c/h in registers.
//   K3: input-projection GEMM for layer 2 [T*B,50]x[50,40], 3 N-tiles/wave.
//   K4: tiny LSTM2 scan (VALU).
//   K5: FC head.
// Out-of-range N columns are computed with clamped (finite) B rows and simply
// not stored; out-of-range K uses zeroed A operands, so 0 * finite = 0.
// ---------------------------------------------------------------------------

typedef float v2f __attribute__((ext_vector_type(2)));
typedef float v8f __attribute__((ext_vector_type(8)));

constexpr int kB  = 256;
constexpr int kT  = 512;
constexpr int kE  = 300;
constexpr int kH1 = 50;
constexpr int kH2 = 10;
constexpr int kG1 = 4 * kH1;   // 200
constexpr int kG2 = 4 * kH2;   // 40

static __device__ __forceinline__ v8f wmma_f32(v2f a, v2f b, v8f c) {
  // (neg_a, A, neg_b, B, c_mod, C, reuse_a, reuse_b)
  return __builtin_amdgcn_wmma_f32_16x16x4_f32(false, a, false, b,
                                               (short)0, c, false, false);
}

static __device__ __forceinline__ float sigmoidf(float x) {
  return 1.0f / (1.0f + __expf(-x));
}

// ---------------------------------------------------------------------------
// K1: gates1x[t*B+b, n] = sum_e emb[x[b,t], e] * W_ih1[n, e] + b1[n]
// M = T*B = 131072 (grid.y, 16 rows/wave), N = 200 -> 2 groups of 7 tiles
// (tile 13 is pure padding), K = 300 (75 steps of 4).
// A-lane layout (16x4 f32): lanes 0-15 -> K = k0+{0,1}; lanes 16-31 -> +{2,3};
// B mirrored; C/D: lane%16 = N, VGPR v = M row v + 8*(lane/16).
// ---------------------------------------------------------------------------
__global__ void __launch_bounds__(32)
k_embed_gemm1(const int* __restrict__ x, const float* __restrict__ emb,
              const float* __restrict__ W_ih1, const float* __restrict__ b1,
              float* __restrict__ gates1x) {
  const int lane = threadIdx.x;
  const int hi   = lane >> 4;
  const int lo   = lane & 15;
  const int koff = hi * 2;
  const int g0   = blockIdx.x;        // N-tile group (0..1), 7 tiles each
  const int m0   = blockIdx.y * 16;

  // A row for this lane (time-major row r = t*B + b; x is [B,T] row-major)
  const int r   = m0 + lo;
  const int t   = r / kB;
  const int b   = r - t * kB;
  const int tok = x[b * kT + t];
  const float* __restrict__ arow = emb + (size_t)tok * kE;

  // Clamped B rows: columns n >= 200 compute garbage that is never stored.
  const float* brow[7];
  int ncol[7];
#pragma unroll
  for (int tt = 0; tt < 7; ++tt) {
    const int n = (g0 * 7 + tt) * 16 + lo;
    ncol[tt]   = n;
    const int nc = (n < kG1) ? n : (kG1 - 1);
    brow[tt]   = W_ih1 + (size_t)nc * kE;
  }

  v8f acc[7];
#pragma unroll
  for (int tt = 0; tt < 7; ++tt) acc[tt] = (v8f){};

  for (int k0 = 0; k0 < kE; k0 += 4) {
    const v2f av = *(const v2f*)(arow + k0 + koff);
#pragma unroll
    for (int tt = 0; tt < 7; ++tt) {
      const v2f bv = *(const v2f*)(brow[tt] + k0 + koff);
      acc[tt] = wmma_f32(av, bv, acc[tt]);
    }
  }

#pragma unroll
  for (int tt = 0; tt < 7; ++tt) {
    const int n = ncol[tt];
    if (n < kG1) {
      const float bias = b1[n];
#pragma unroll
      for (int v = 0; v < 8; ++v) {
        const int m = m0 + v + 8 * hi;
        gates1x[(size_t)m * kG1 + n] = acc[tt][v] + bias;
      }
    }
  }
}

// ---------------------------------------------------------------------------
// K3: gates2x[r, n] = sum_k hs1[r, k] * W_ih2[n, k] + b2[n]
// M = 131072 (grid.x), N = 40 (3 tiles, all in one wave), K = 50:
// 12 full k-steps + 1 tail step with the hi half-wave's A zero-selected.
// ---------------------------------------------------------------------------
__global__ void __launch_bounds__(32)
k_gemm2(const float* __restrict__ hs1, const float* __restrict__ W_ih2,
        const float* __restrict__ b2, float* __restrict__ gates2x) {
  const int lane = threadIdx.x;
  const int hi   = lane >> 4;
  const int lo   = lane & 15;
  const int koff = hi * 2;
  const int m0   = blockIdx.x * 16;

  const int r = m0 + lo;
  const float* __restrict__ arow = hs1 + (size_t)r * kH1;

  const float* brow[3];
  int ncol[3];
#pragma unroll
  for (int tt = 0; tt < 3; ++tt) {
    const int n = tt * 16 + lo;
    ncol[tt]   = n;
    const int nc = (n < kG2) ? n : (kG2 - 1);
    brow[tt]   = W_ih2 + (size_t)nc * kH1;
  }

  v8f acc[3];
#pragma unroll
  for (int tt = 0; tt < 3; ++tt) acc[tt] = (v8f){};

#pragma unroll
  for (int k0 = 0; k0 < 48; k0 += 4) {
    const v2f av = *(const v2f*)(arow + k0 + koff);
#pragma unroll
    for (int tt = 0; tt < 3; ++tt) {
      const v2f bv = *(const v2f*)(brow[tt] + k0 + koff);
      acc[tt] = wmma_f32(av, bv, acc[tt]);
    }
  }
  {
    // Tail: K = 48..51.  All lanes load k={48,49}; hi lanes (logical K=50,51)
    // zero-select A so their (finite) B values contribute exactly zero.
    const v2f z2 = {};
    v2f av = *(const v2f*)(arow + 48);
    av = hi ? z2 : av;
#pragma unroll
    for (int tt = 0; tt < 3; ++tt) {
      const v2f bv = *(const v2f*)(brow[tt] + 48);
      acc[tt] = wmma_f32(av, bv, acc[tt]);
    }
  }

#pragma unroll
  for (int tt = 0; tt < 3; ++tt) {
    const int n = ncol[tt];
    if (n < kG2) {
      const float bias = b2[n];
#pragma unroll
      for (int v = 0; v < 8; ++v) {
        const int m = m0 + v + 8 * hi;
        gates2x[(size_t)m * kG2 + n] = acc[tt][v] + bias;
      }
    }
  }
}

// ---------------------------------------------------------------------------
// K2: LSTM layer-1 persistent scan.  16 blocks x 16 batch rows, 256 threads.
// Per step: gates(16x200) = gates1x(t) + h(16x50) @ W_hh1^T via WMMA from LDS;
// then elementwise LSTM cell with c/h held in registers (fixed ownership).
// Masked (padded) steps freeze h and c, matching the reference semantics.
// ---------------------------------------------------------------------------
__global__ void __launch_bounds__(256)
k_lstm1_scan(const float* __restrict__ gates1x,
             const float* __restrict__ W_hh1,
             const int* __restrict__ lengths,
             float* __restrict__ hs1) {
  __shared__ float W_lds[208 * 52];  // [n][k], zero-padded n>=200, k>=50
  __shared__ float h_lds[16 * 52];   // [row][k], zero-padded k>=50
  __shared__ float g_lds[16 * 208];  // [row][n]; cols >=200 never read

  const int tid  = threadIdx.x;
  const int b0   = blockIdx.x * 16;
  const int lane = tid & 31;
  const int wave = tid >> 5;
  const int hi   = lane >> 4;
  const int lo   = lane & 15;
  const int koff = hi * 2;

  for (int i = tid; i < 208 * 52; i += 256) {
    const int nn = i / 52, kk = i - nn * 52;
    W_lds[i] = (nn < kG1 && kk < kH1) ? W_hh1[nn * kH1 + kk] : 0.0f;
  }
  for (int i = tid; i < 16 * 52; i += 256) h_lds[i] = 0.0f;

  // Fixed per-thread cell ownership: cells j = tid + 256*u, j < 800.
  float creg[4], hreg[4];
  int   rowc[4], colc[4], lenc[4];
#pragma unroll
  for (int u = 0; u < 4; ++u) {
    const int j = tid + u * 256;
    creg[u] = 0.0f;
    hreg[u] = 0.0f;
    if (j < 16 * kH1) {
      rowc[u] = j / kH1;
      colc[u] = j - rowc[u] * kH1;
      lenc[u] = lengths[b0 + rowc[u]];
    } else {
      rowc[u] = -1; colc[u] = 0; lenc[u] = 0;
    }
  }
  __syncthreads();

  for (int t = 0; t < kT; ++t) {
    // ---- recurrent GEMM: 13 N-tiles, wave w handles tiles w, w+8 ----
    const size_t gbase = ((size_t)t * kB + b0) * kG1;
    for (int tile = wave; tile < 13; tile += 8) {
      const int n   = tile * 16 + lo;
      const int ncl = (n < kG1) ? n : (kG1 - 1);  // branch-free C init
      v8f acc;
#pragma unroll
      for (int v = 0; v < 8; ++v) {
        const int m = v + 8 * hi;
        acc[v] = gates1x[gbase + (size_t)m * kG1 + ncl];
      }
#pragma unroll
      for (int k0 = 0; k0 < 52; k0 += 4) {
        const v2f av = *(const v2f*)(&h_lds[lo * 52 + k0 + koff]);
        const v2f bv = *(const v2f*)(&W_lds[n * 52 + k0 + koff]);
        acc = wmma_f32(av, bv, acc);
      }
#pragma unroll
      for (int v = 0; v < 8; ++v)
        g_lds[(v + 8 * hi) * 208 + n] = acc[v];
    }
    __syncthreads();

    // ---- elementwise LSTM cell (gate order i,f,g,o) ----
#pragma unroll
    for (int u = 0; u < 4; ++u) {
      if (rowc[u] >= 0) {
        const int row = rowc[u], nn = colc[u];
        const float ig = g_lds[row * 208 + nn];
        const float fg = g_lds[row * 208 + kH1 + nn];
        const float gg = g_lds[row * 208 + 2 * kH1 + nn];
        const float og = g_lds[row * 208 + 3 * kH1 + nn];
        const float cn = sigmoidf(fg) * creg[u] + sigmoidf(ig) * tanhf(gg);
        const float hn = sigmoidf(og) * tanhf(cn);
        const bool valid = (t < lenc[u]);
        creg[u] = valid ? cn : creg[u];
        hreg[u] = valid ? hn : hreg[u];
        h_lds[row * 52 + nn] = hreg[u];
        hs1[((size_t)t * kB + b0 + row) * kH1 + nn] = hreg[u];
      }
    }
    __syncthreads();
  }
}

// ---------------------------------------------------------------------------
// K4: LSTM layer-2 scan (tiny: H=10).  16 blocks x 16 rows, 256 threads.
// Writes only the final hidden state (padded steps freeze state == hT).
// ---------------------------------------------------------------------------
__global__ void __launch_bounds__(256)
k_lstm2_scan(const float* __restrict__ gates2x,
             const float* __restrict__ W_hh2,
             const int* __restrict__ lengths,
             float* __restrict__ h2out) {
  __shared__ float W2[kG2 * kH2];     // 400
  __shared__ float h_lds[16 * kH2];   // 160
  __shared__ float g_lds[16 * kG2];   // 640

  const int tid = threadIdx.x;
  const int b0  = blockIdx.x * 16;

  for (int i = tid; i < kG2 * kH2; i += 256) W2[i] = W_hh2[i];
  for (int i = tid; i < 16 * kH2; i += 256) h_lds[i] = 0.0f;

  float creg = 0.0f, hreg = 0.0f;
  int myrow = -1, mycol = 0, mylen = 0;
  if (tid < 16 * kH2) {
    myrow = tid / kH2;
    mycol = tid - myrow * kH2;
    mylen = lengths[b0 + myrow];
  }
  __syncthreads();

  for (int t = 0; t < kT; ++t) {
    const size_t gbase = ((size_t)t * kB + b0) * kG2;
    for (int j = tid; j < 16 * kG2; j += 256) {
      const int row = j / kG2, nn = j - row * kG2;
      float g = gates2x[gbase + (size_t)row * kG2 + nn];
#pragma unroll
      for (int k = 0; k < kH2; ++k) g += h_lds[row * kH2 + k] * W2[nn * kH2 + k];
      g_lds[j] = g;
    }
    __syncthreads();
    if (myrow >= 0) {
      const float ig = g_lds[myrow * kG2 + mycol];
      const float fg = g_lds[myrow * kG2 + kH2 + mycol];
      const float gg = g_lds[myrow * kG2 + 2 * kH2 + mycol];
      const float og = g_lds[myrow * kG2 + 3 * kH2 + mycol];
      const float cn = sigmoidf(fg) * creg + sigmoidf(ig) * tanhf(gg);
      const float hn = sigmoidf(og) * tanhf(cn);
      const bool valid = (t < mylen);
      creg = valid ? cn : creg;
      hreg = valid ? hn : hreg;
      h_lds[myrow * kH2 + mycol] = hreg;
    }
    __syncthreads();
  }
  if (myrow >= 0) h2out[(b0 + myrow) * kH2 + mycol] = hreg;
}

// ---------------------------------------------------------------------------
// K5: out[b] = fc2_w . relu(fc1_w @ h2[b] + fc1_b) + fc2_b
// ---------------------------------------------------------------------------
__global__ void __launch_bounds__(256)
k_head(const float* __restrict__ h2, const float* __restrict__ fc1_w,
       const float* __restrict__ fc1_b, const float* __restrict__ fc2_w,
       const float* __restrict__ fc2_b, float* __restrict__ out) {
  const int b = blockIdx.x * blockDim.x + threadIdx.x;
  if (b >= kB) return;
  float h[kH2];
#pragma unroll
  for (int k = 0; k < kH2; ++k) h[k] = h2[b * kH2 + k];
  float o = fc2_b[0];
#pragma unroll
  for (int j = 0; j < 10; ++j) {
    float z = fc1_b[j];
#pragma unroll
    for (int k = 0; k < kH2; ++k) z += fc1_w[j * kH2 + k] * h[k];
    o += fc2_w[j] * fmaxf(z, 0.0f);
  }
  out[b] = o;
}

// ---------------------------------------------------------------------------
extern "C" void kernel_launch(void* const* d_in, const int* in_sizes, int n_in,
                              void* d_out, int out_size, void* d_ws,
                              size_t ws_size, hipStream_t stream) {
  const int*   x       = (const int*)d_in[0];
  const int*   lengths = (const int*)d_in[1];
  const float* emb     = (const float*)d_in[2];
  const float* W_ih1   = (const float*)d_in[3];
  const float* W_hh1   = (const float*)d_in[4];
  const float* b1      = (const float*)d_in[5];
  const float* W_ih2   = (const float*)d_in[6];
  const float* W_hh2   = (const float*)d_in[7];
  const float* b2      = (const float*)d_in[8];
  const float* fc1_w   = (const float*)d_in[9];
  const float* fc1_b   = (const float*)d_in[10];
  const float* fc2_w   = (const float*)d_in[11];
  const float* fc2_b   = (const float*)d_in[12];
  float*       out     = (float*)d_out;

  // Workspace layout (floats): gates1x | hs1 | gates2x | h2
  float* ws      = (float*)d_ws;
  float* gates1x = ws;                                    // T*B*200
  float* hs1     = gates1x + (size_t)kT * kB * kG1;       // T*B*50
  float* gates2x = hs1     + (size_t)kT * kB * kH1;       // T*B*40
  float* h2buf   = gates2x + (size_t)kT * kB * kG2;       // B*10

  const int mtiles = (kT * kB) / 16;  // 8192

  dim3 g1(2, mtiles);                 // 2 groups of 7 N-tiles
  k_embed_gemm1<<<g1, 32, 0, stream>>>(x, emb, W_ih1, b1, gates1x);

  k_lstm1_scan<<<kB / 16, 256, 0, stream>>>(gates1x, W_hh1, lengths, hs1);

  k_gemm2<<<mtiles, 32, 0, stream>>>(hs1, W_ih2, b2, gates2x);

  k_lstm2_scan<<<kB / 16, 256, 0, stream>>>(gates2x, W_hh2, lengths, h2buf);

  k_head<<<1, 256, 0, stream>>>(h2buf, fc1_w, fc1_b, fc2_w, fc2_b, out);
}